// MoE_77721728189053
// MI455X (gfx1250) — compile-verified
//
#include <hip/hip_runtime.h>
#include <stddef.h>
#include <stdint.h>

// Problem constants (match reference)
constexpr int TOK  = 4096;   // T
constexpr int DIM  = 1024;   // D
constexpr int IDIM = 1024;   // I
constexpr int NE   = 8;      // experts
// TOPK = 2 (hardcoded in gate kernel)

// GEMM tiling
constexpr int TM  = 64;   // token rows per block tile
constexpr int TN  = 64;   // output cols per block tile
constexpr int TK  = 16;   // K step (4 wmma sub-steps of k=4)
constexpr int TKP = 20;   // padded K stride: 16B-aligned b128 stores,
                          // conflict-free b64 fragment reads (64 banks)

typedef __attribute__((ext_vector_type(2))) float v2f;
typedef __attribute__((ext_vector_type(8))) float v8f;

// ---------------------------------------------------------------------------
// gfx1250 async copy: global -> LDS, 16B per lane, tracked by ASYNCcnt
// ---------------------------------------------------------------------------
__device__ __forceinline__ unsigned lds_lo32(const void* p) {
    return (unsigned)(uintptr_t)p;   // low 32 bits of generic ptr = LDS byte addr
}
__device__ __forceinline__ void async_b128(unsigned lds, const float* g) {
    asm volatile("global_load_async_to_lds_b128 %0, %1, off"
                 :: "v"(lds), "v"(g) : "memory");
}
__device__ __forceinline__ void wait_async0() {
    asm volatile("s_wait_asynccnt 0x0" ::: "memory");
}

// ---------------------------------------------------------------------------
// wave-level f32 WMMA (16x16x4): D = A(16x4) * B(4x16) + C
// A frag: lanes 0-15 -> row=lane, K={k,k+1}; lanes 16-31 -> row=lane-16, K={k+2,k+3}
// B frag: lanes 0-15 -> col=lane, K={k,k+1}; lanes 16-31 -> col=lane-16, K={k+2,k+3}
// C/D   : VGPR r -> M = r + 8*(lane>=16), N = lane%16
// ---------------------------------------------------------------------------
__device__ __forceinline__ v8f wmma_f32(v2f a, v2f b, v8f c) {
    return __builtin_amdgcn_wmma_f32_16x16x4_f32(
        false, a, false, b, (short)0, c, false, false);
}

// ---------------------------------------------------------------------------
// 0) zero y + expert counters
// ---------------------------------------------------------------------------
__global__ void zero_kernel(float* __restrict__ y, int* __restrict__ counts, int n) {
    for (int i = blockIdx.x * blockDim.x + threadIdx.x; i < n;
         i += gridDim.x * blockDim.x)
        y[i] = 0.0f;
    if (blockIdx.x == 0 && threadIdx.x < NE) counts[threadIdx.x] = 0;
}

// ---------------------------------------------------------------------------
// 1) gating: softmax over 8 experts, top-2, append to per-expert queues
// ---------------------------------------------------------------------------
__global__ __launch_bounds__(128)
void gate_topk_kernel(const float* __restrict__ x,
                      const float* __restrict__ gw,
                      int*  __restrict__ counts,
                      int*  __restrict__ tokq,
                      float* __restrict__ wtq) {
    const int lane = threadIdx.x & 31;
    const int wv   = threadIdx.x >> 5;
    const int t    = blockIdx.x * 4 + wv;
    if (t >= TOK) return;

    float acc[NE];
#pragma unroll
    for (int e = 0; e < NE; ++e) acc[e] = 0.0f;

    const float* xr = x + (size_t)t * DIM;
    for (int d = lane; d < DIM; d += 32) {
        const float xv = xr[d];
#pragma unroll
        for (int e = 0; e < NE; ++e)
            acc[e] = fmaf(xv, gw[e * DIM + d], acc[e]);
    }
#pragma unroll
    for (int e = 0; e < NE; ++e) {
#pragma unroll
        for (int off = 16; off > 0; off >>= 1)
            acc[e] += __shfl_xor(acc[e], off, 32);
    }
    float mx = acc[0];
#pragma unroll
    for (int e = 1; e < NE; ++e) mx = fmaxf(mx, acc[e]);
    float p[NE], sum = 0.0f;
#pragma unroll
    for (int e = 0; e < NE; ++e) { p[e] = __expf(acc[e] - mx); sum += p[e]; }
    const float inv = 1.0f / sum;
#pragma unroll
    for (int e = 0; e < NE; ++e) p[e] *= inv;

    int i1 = 0; float v1 = p[0];
#pragma unroll
    for (int e = 1; e < NE; ++e) if (p[e] > v1) { v1 = p[e]; i1 = e; }
    int i2 = (i1 == 0) ? 1 : 0; float v2 = p[i2];
#pragma unroll
    for (int e = 0; e < NE; ++e)
        if (e != i1 && p[e] > v2) { v2 = p[e]; i2 = e; }

    if (lane == 0) {
        int s1 = atomicAdd(&counts[i1], 1);
        tokq[i1 * TOK + s1] = t;  wtq[i1 * TOK + s1] = v1;
        int s2 = atomicAdd(&counts[i2], 1);
        tokq[i2 * TOK + s2] = t;  wtq[i2 * TOK + s2] = v2;
    }
}

// ---------------------------------------------------------------------------
// 2) exclusive scan of 8 counters -> act-row offsets
// ---------------------------------------------------------------------------
__global__ void scan_kernel(const int* __restrict__ counts, int* __restrict__ offsets) {
    if (threadIdx.x == 0 && blockIdx.x == 0) {
        int s = 0;
#pragma unroll
        for (int e = 0; e < NE; ++e) { offsets[e] = s; s += counts[e]; }
        offsets[NE] = s;
    }
}

// ---------------------------------------------------------------------------
// 3) expert up-projection: h = Xg@w1^T, g = Xg@w3^T, act = silu(h)*g
//    8 waves; wave = 32 rows x 16 cols; double-buffered async LDS tiles
// ---------------------------------------------------------------------------
__global__ __launch_bounds__(256)
void expert_up_kernel(const float* __restrict__ x,
                      const float* __restrict__ w1,
                      const float* __restrict__ w3,
                      const int*  __restrict__ counts,
                      const int*  __restrict__ offsets,
                      const int*  __restrict__ tokq,
                      float* __restrict__ act) {
    const int e   = blockIdx.x / (TOK / TM);
    const int mt  = blockIdx.x % (TOK / TM);
    const int cnt = counts[e];
    const int m0  = mt * TM;
    if (m0 >= cnt) return;                 // block-uniform exit
    const int n0   = blockIdx.y * TN;
    const int base = offsets[e];

    __shared__ __align__(16) float As [2][TM][TKP];   // A[m][k]
    __shared__ __align__(16) float B1s[2][TN][TKP];   // B[n][k]
    __shared__ __align__(16) float B3s[2][TN][TKP];
    __shared__ int toks[TM];

    const int tid  = threadIdx.x;
    const int lane = tid & 31;
    const int wv   = tid >> 5;
    const int hlf  = lane >> 4;
    const int lr   = lane & 15;
    const int m0w  = (wv & 1) * 32;        // wave rows: m0w..m0w+31
    const int nc   = (wv >> 1) * 16;       // wave cols: nc..nc+15

    const int lm = tid >> 2;               // 0..63: staged row
    const int lk = (tid & 3) * 4;          // 0,4,8,12: staged k-quad

    if (tid < TM) {
        const int slot = m0 + tid;
        toks[tid] = tokq[e * TOK + (slot < cnt ? slot : cnt - 1)]; // clamp: rows
    }                                                              // >=cnt never stored
    __syncthreads();

    const float* __restrict__ xrow = x  + (size_t)toks[lm] * DIM;
    const float* __restrict__ w1r  = w1 + ((size_t)e * IDIM + n0 + lm) * DIM;
    const float* __restrict__ w3r  = w3 + ((size_t)e * IDIM + n0 + lm) * DIM;

    // prologue: tile 0 -> buffer 0
    async_b128(lds_lo32(&As [0][lm][lk]), xrow + lk);
    async_b128(lds_lo32(&B1s[0][lm][lk]), w1r  + lk);
    async_b128(lds_lo32(&B3s[0][lm][lk]), w3r  + lk);

    v8f cH0 = {}, cH1 = {}, cG0 = {}, cG1 = {};

    constexpr int NT = DIM / TK;
    for (int t = 0; t < NT; ++t) {
        wait_async0();                      // my tile-t copies landed
        __syncthreads();                    // everyone's copies visible
        if (t + 1 < NT) {                   // overlap tile t+1 with compute
            const int k0n = (t + 1) * TK;
            const int s1  = (t + 1) & 1;
            async_b128(lds_lo32(&As [s1][lm][lk]), xrow + k0n + lk);
            async_b128(lds_lo32(&B1s[s1][lm][lk]), w1r  + k0n + lk);
            async_b128(lds_lo32(&B3s[s1][lm][lk]), w3r  + k0n + lk);
        }
        const int s = t & 1;
#pragma unroll
        for (int kk = 0; kk < TK; kk += 4) {
            const int ka = kk + 2 * hlf;
            const v2f a0 = *(const v2f*)&As [s][m0w + lr][ka];
            const v2f a1 = *(const v2f*)&As [s][m0w + 16 + lr][ka];
            const v2f b1 = *(const v2f*)&B1s[s][nc + lr][ka];
            const v2f b3 = *(const v2f*)&B3s[s][nc + lr][ka];
            cH0 = wmma_f32(a0, b1, cH0);
            cH1 = wmma_f32(a1, b1, cH1);
            cG0 = wmma_f32(a0, b3, cG0);
            cG1 = wmma_f32(a1, b3, cG1);
        }
    }

    // epilogue: act = silu(h) * g   (guarded rows only)
    const int col = n0 + nc + lr;
#pragma unroll
    for (int r = 0; r < 8; ++r) {
        const int slot0 = m0 + m0w + r + 8 * hlf;
        if (slot0 < cnt) {
            const float h = cH0[r];
            act[(size_t)(base + slot0) * IDIM + col] =
                (h / (1.0f + __expf(-h))) * cG0[r];
        }
        const int slot1 = slot0 + 16;
        if (slot1 < cnt) {
            const float h = cH1[r];
            act[(size_t)(base + slot1) * IDIM + col] =
                (h / (1.0f + __expf(-h))) * cG1[r];
        }
    }
}

// ---------------------------------------------------------------------------
// 4) expert down-projection + weighted scatter: y[t] += wt * (act @ w2^T)
// ---------------------------------------------------------------------------
__global__ __launch_bounds__(256)
void expert_down_kernel(const float* __restrict__ act,
                        const float* __restrict__ w2,
                        const int*  __restrict__ counts,
                        const int*  __restrict__ offsets,
                        const int*  __restrict__ tokq,
                        const float* __restrict__ wtq,
                        float* __restrict__ y) {
    const int e   = blockIdx.x / (TOK / TM);
    const int mt  = blockIdx.x % (TOK / TM);
    const int cnt = counts[e];
    const int m0  = mt * TM;
    if (m0 >= cnt) return;
    const int n0   = blockIdx.y * TN;      // over DIM
    const int base = offsets[e];

    __shared__ __align__(16) float As[2][TM][TKP];
    __shared__ __align__(16) float Bs[2][TN][TKP];
    __shared__ int   toks[TM];
    __shared__ float wts[TM];

    const int tid  = threadIdx.x;
    const int lane = tid & 31;
    const int wv   = tid >> 5;
    const int hlf  = lane >> 4;
    const int lr   = lane & 15;
    const int m0w  = (wv & 1) * 32;
    const int nc   = (wv >> 1) * 16;

    const int lm = tid >> 2;
    const int lk = (tid & 3) * 4;

    if (tid < TM) {
        const int slot  = m0 + tid;
        const int slotc = slot < cnt ? slot : cnt - 1;
        toks[tid] = tokq[e * TOK + slotc];
        wts[tid]  = wtq [e * TOK + slotc];
    }
    __syncthreads();

    const int rowc = (m0 + lm < cnt) ? (m0 + lm) : (cnt - 1);   // clamp ragged rows
    const float* __restrict__ arow = act + (size_t)(base + rowc) * IDIM;
    const float* __restrict__ w2r  = w2 + ((size_t)e * DIM + n0 + lm) * IDIM;

    async_b128(lds_lo32(&As[0][lm][lk]), arow + lk);
    async_b128(lds_lo32(&Bs[0][lm][lk]), w2r  + lk);

    v8f c0 = {}, c1 = {};

    constexpr int NT = IDIM / TK;
    for (int t = 0; t < NT; ++t) {
        wait_async0();
        __syncthreads();
        if (t + 1 < NT) {
            const int k0n = (t + 1) * TK;
            const int s1  = (t + 1) & 1;
            async_b128(lds_lo32(&As[s1][lm][lk]), arow + k0n + lk);
            async_b128(lds_lo32(&Bs[s1][lm][lk]), w2r  + k0n + lk);
        }
        const int s = t & 1;
#pragma unroll
        for (int kk = 0; kk < TK; kk += 4) {
            const int ka = kk + 2 * hlf;
            const v2f a0 = *(const v2f*)&As[s][m0w + lr][ka];
            const v2f a1 = *(const v2f*)&As[s][m0w + 16 + lr][ka];
            const v2f b  = *(const v2f*)&Bs[s][nc + lr][ka];
            c0 = wmma_f32(a0, b, c0);
            c1 = wmma_f32(a1, b, c1);
        }
    }

    // epilogue: y[token] += routing_weight * tile  (2 commutative adds / element)
    const int col = n0 + nc + lr;
#pragma unroll
    for (int r = 0; r < 8; ++r) {
        const int mrow0 = m0w + r + 8 * hlf;
        if (m0 + mrow0 < cnt)
            atomicAdd(&y[(size_t)toks[mrow0] * DIM + col], c0[r] * wts[mrow0]);
        const int mrow1 = mrow0 + 16;
        if (m0 + mrow1 < cnt)
            atomicAdd(&y[(size_t)toks[mrow1] * DIM + col], c1[r] * wts[mrow1]);
    }
}

// ---------------------------------------------------------------------------
// launch: zero -> gate/top2 -> scan -> up(silu) -> down(scatter)
// workspace: [counts][offsets][tokq E*T*4][wtq E*T*4][act 2T*I*4] ~ 34 MB
// ---------------------------------------------------------------------------
extern "C" void kernel_launch(void* const* d_in, const int* in_sizes, int n_in,
                              void* d_out, int out_size, void* d_ws, size_t ws_size,
                              hipStream_t stream) {
    (void)in_sizes; (void)n_in; (void)out_size; (void)ws_size;
    const float* x  = (const float*)d_in[0];
    const float* gw = (const float*)d_in[1];
    const float* w1 = (const float*)d_in[2];
    const float* w3 = (const float*)d_in[3];
    const float* w2 = (const float*)d_in[4];
    float* y = (float*)d_out;

    char* ws = (char*)d_ws;
    int*   counts  = (int*)(ws);
    int*   offsets = (int*)(ws + 64);
    int*   tokq    = (int*)(ws + 256);
    float* wtq     = (float*)(ws + 256 + (size_t)NE * TOK * 4);
    float* act     = (float*)(ws + 256 + (size_t)2 * NE * TOK * 4);

    zero_kernel<<<1024, 256, 0, stream>>>(y, counts, TOK * DIM);
    gate_topk_kernel<<<TOK / 4, 128, 0, stream>>>(x, gw, counts, tokq, wtq);
    scan_kernel<<<1, 32, 0, stream>>>(counts, offsets);

    dim3 gUp(NE * (TOK / TM), IDIM / TN);
    expert_up_kernel<<<gUp, 256, 0, stream>>>(x, w1, w3, counts, offsets, tokq, act);

    dim3 gDn(NE * (TOK / TM), DIM / TN);
    expert_down_kernel<<<gDn, 256, 0, stream>>>(act, w2, counts, offsets, tokq, wtq, y);
}